// MlaQKVGemm_79216376808022
// MI455X (gfx1250) — compile-verified
//
#include <hip/hip_runtime.h>
#include <hip/hip_bf16.h>

// ---------------- MLA QKV projection for MI455X (gfx1250, wave32) ----------
// 4 fp32 GEMMs on V_WMMA_F32_16X16X4_F32 + fused rmsnorm / rope-shuffle
// epilogues. Tile: 128(M) x 64(N) per 256-thread block (8 waves), each wave
// owns a 16x64 slab = 4 x v8f accumulators.
// Copy path: GLOBAL_LOAD_ASYNC_TO_LDS_B128 + ASYNCcnt double buffering when
// the builtins exist; otherwise VGPR-staged sync path.

typedef float v2f __attribute__((ext_vector_type(2)));
typedef float v4f __attribute__((ext_vector_type(4)));
typedef float v8f __attribute__((ext_vector_type(8)));
typedef int   i32x4 __attribute__((vector_size(16)));   // matches builtin proto

#define BM 128
#define BN 64
#define BK 16
#define AS_STRIDE 20   // 16 + 4 pad: float4-aligned, conflict-free frag reads
#define BS_STRIDE 20
#define A_TILE (BM * AS_STRIDE)
#define B_TILE (BN * BS_STRIDE)

#if defined(__has_builtin)
#if __has_builtin(__builtin_amdgcn_global_load_async_to_lds_b128) && \
    __has_builtin(__builtin_amdgcn_s_wait_asynccnt)
#define HAS_ASYNC 1
#endif
#endif
#ifndef HAS_ASYNC
#define HAS_ASYNC 0
#endif

#if HAS_ASYNC
typedef __attribute__((address_space(1))) i32x4 gi32x4;  // global int4
typedef __attribute__((address_space(3))) i32x4 li32x4;  // LDS int4

__device__ __forceinline__ void async_b128(const float* g, float* l) {
    // AS1 and flat share the VA for global memory -> int round-trip is exact.
    // LDS side is a real addrspacecast (explicit cast flat -> local).
    __builtin_amdgcn_global_load_async_to_lds_b128(
        (gi32x4*)(uintptr_t)g,
        (li32x4*)l,
        0, 0);
}

// Issue the 3 async b128 copies for one K-chunk (per thread).
__device__ __forceinline__ void stage_issue(
    const float* __restrict__ A, const float* __restrict__ W, int K,
    int rowA0, int nW0, int k0, float* As, float* Bs, int tid)
{
#pragma unroll
    for (int i = 0; i < 2; ++i) {
        int f4id = tid * 2 + i;            // 0..511
        int r  = f4id >> 2;
        int kq = f4id & 3;
        async_b128(A + (size_t)(rowA0 + r) * K + k0 + kq * 4,
                   As + r * AS_STRIDE + kq * 4);
    }
    {
        int n  = tid >> 2;
        int kq = tid & 3;
        async_b128(W + (size_t)(nW0 + n) * K + k0 + kq * 4,
                   Bs + n * BS_STRIDE + kq * 4);
    }
}
#endif

// 16 WMMAs on one staged K-chunk.
__device__ __forceinline__ void compute_chunk(
    const float* __restrict__ As, const float* __restrict__ Bs,
    int mloc, int l, int half, v8f acc[4])
{
#pragma unroll
    for (int ks = 0; ks < BK; ks += 4) {
        // A 16x4 layout: lanes 0-15 hold K=0,1 (VGPR0,1); lanes 16-31 K=2,3
        v2f a = *(const v2f*)(As + (mloc + l) * AS_STRIDE + ks + 2 * half);
#pragma unroll
        for (int t = 0; t < 4; ++t) {
            v2f b = *(const v2f*)(Bs + (t * 16 + l) * BS_STRIDE + ks + 2 * half);
            acc[t] = __builtin_amdgcn_wmma_f32_16x16x4_f32(
                false, a, false, b, (short)0, acc[t], false, false);
        }
    }
}

// Core fp32 WMMA mainloop: C[128x64] += A[rowA0.., K] * W[nW0.., K]^T
// A is MxK row-major, W is NxK row-major (we compute X @ W^T).
__device__ __forceinline__ void gemm_tile_f32(
    const float* __restrict__ A, const float* __restrict__ W,
    int K, int rowA0, int nW0,
    float* __restrict__ As, float* __restrict__ Bs, v8f acc[4])
{
    const int tid  = threadIdx.x;
    const int lane = tid & 31;
    const int wave = tid >> 5;      // 0..7 -> M sub-row of tile
    const int l    = lane & 15;
    const int half = lane >> 4;
    const int mloc = wave * 16;

    v8f z = {};
#pragma unroll
    for (int t = 0; t < 4; ++t) acc[t] = z;

#if HAS_ASYNC
    const int nch = K / BK;
    stage_issue(A, W, K, rowA0, nW0, 0, As, Bs, tid);
    for (int c = 0; c < nch; ++c) {
        const int cur = c & 1;
        if (c + 1 < nch) {
            stage_issue(A, W, K, rowA0, nW0, (c + 1) * BK,
                        As + (cur ^ 1) * A_TILE, Bs + (cur ^ 1) * B_TILE, tid);
            __builtin_amdgcn_s_wait_asynccnt(3);   // chunk c landed
        } else {
            __builtin_amdgcn_s_wait_asynccnt(0);
        }
        __syncthreads();
        compute_chunk(As + cur * A_TILE, Bs + cur * B_TILE, mloc, l, half, acc);
        __syncthreads();   // protect buf reuse two iterations out
    }
#else
    for (int k0 = 0; k0 < K; k0 += BK) {
        // ---- stage A tile: 128x16 floats, 2 contiguous float4 per thread
#pragma unroll
        for (int i = 0; i < 2; ++i) {
            int f4id = tid * 2 + i;            // 0..511
            int r  = f4id >> 2;
            int kq = f4id & 3;
            v4f v = *(const v4f*)(A + (size_t)(rowA0 + r) * K + k0 + kq * 4);
            *(v4f*)(As + r * AS_STRIDE + kq * 4) = v;
        }
        // ---- stage B tile as Bs[n][k]: 64x16 floats, 1 float4 per thread
        {
            int n  = tid >> 2;
            int kq = tid & 3;
            v4f v = *(const v4f*)(W + (size_t)(nW0 + n) * K + k0 + kq * 4);
            *(v4f*)(Bs + n * BS_STRIDE + kq * 4) = v;
        }
        __syncthreads();
        compute_chunk(As, Bs, mloc, l, half, acc);
        __syncthreads();
    }
#endif
}

// ---------------- GEMM 1: hidden @ q_a_w^T -> q_mid[8192 x 1536] -----------
__global__ void __launch_bounds__(256) gemm_qa(
    const float* __restrict__ X, const float* __restrict__ W,
    float* __restrict__ q_mid)
{
    __shared__ float As[2 * A_TILE];
    __shared__ float Bs[2 * B_TILE];
    v8f acc[4];
    const int rowA0 = blockIdx.y * BM;
    const int nW0   = blockIdx.x * BN;
    gemm_tile_f32(X, W, 4096, rowA0, nW0, As, Bs, acc);

    const int lane = threadIdx.x & 31, wave = threadIdx.x >> 5;
    const int l = lane & 15, half = lane >> 4;
    const int row = rowA0 + wave * 16 + half * 8;
#pragma unroll
    for (int t = 0; t < 4; ++t)
#pragma unroll
        for (int v = 0; v < 8; ++v)
            q_mid[(size_t)(row + v) * 1536 + nW0 + t * 16 + l] = acc[t][v];
}

// ---------------- GEMM 2: hidden @ kv_a_w^T -> ckv[8192 x 576] -------------
__global__ void __launch_bounds__(256) gemm_kva(
    const float* __restrict__ X, const float* __restrict__ W,
    float* __restrict__ ckv)
{
    __shared__ float As[2 * A_TILE];
    __shared__ float Bs[2 * B_TILE];
    v8f acc[4];
    const int rowA0 = blockIdx.y * BM;
    const int nW0   = blockIdx.x * BN;
    gemm_tile_f32(X, W, 4096, rowA0, nW0, As, Bs, acc);

    const int lane = threadIdx.x & 31, wave = threadIdx.x >> 5;
    const int l = lane & 15, half = lane >> 4;
    const int row = rowA0 + wave * 16 + half * 8;
#pragma unroll
    for (int t = 0; t < 4; ++t)
#pragma unroll
        for (int v = 0; v < 8; ++v)
            ckv[(size_t)(row + v) * 576 + nW0 + t * 16 + l] = acc[t][v];
}

// ---------------- GEMM 3: q_mid_n @ q_b_w^T -> query part of out -----------
// out channel c = h (0..31); rope dims get deepseek shuffle scatter.
__global__ void __launch_bounds__(256) gemm_qb(
    const float* __restrict__ Qn, const float* __restrict__ W,
    float* __restrict__ out)
{
    __shared__ float As[2 * A_TILE];
    __shared__ float Bs[2 * B_TILE];
    v8f acc[4];
    const int rowA0 = blockIdx.y * BM;
    const int nW0   = blockIdx.x * BN;
    gemm_tile_f32(Qn, W, 1536, rowA0, nW0, As, Bs, acc);

    const int lane = threadIdx.x & 31, wave = threadIdx.x >> 5;
    const int l = lane & 15, half = lane >> 4;
    const int row = rowA0 + wave * 16 + half * 8;
#pragma unroll
    for (int t = 0; t < 4; ++t) {
        const int n = nW0 + t * 16 + l;
        const int h = n / 192;
        const int d = n - h * 192;
        int dd;
        if (d < 128) dd = d;
        else { int j = d - 128; dd = 128 + ((j & 1) * 32 + (j >> 1)); }
#pragma unroll
        for (int v = 0; v < 8; ++v) {
            const int r = row + v;
            const int b = r >> 12, s = r & 4095;
            out[((size_t)(b * 96 + h) * 4096 + s) * 192 + dd] = acc[t][v];
        }
    }
}

// ---------------- GEMM 4: ckv_n @ kv_b_w^T -> k_nope / value parts ---------
__global__ void __launch_bounds__(256) gemm_kvb(
    const float* __restrict__ Cn, const float* __restrict__ W,
    float* __restrict__ out)
{
    __shared__ float As[2 * A_TILE];
    __shared__ float Bs[2 * B_TILE];
    v8f acc[4];
    const int rowA0 = blockIdx.y * BM;
    const int nW0   = blockIdx.x * BN;
    gemm_tile_f32(Cn, W, 512, rowA0, nW0, As, Bs, acc);

    const int lane = threadIdx.x & 31, wave = threadIdx.x >> 5;
    const int l = lane & 15, half = lane >> 4;
    const int row = rowA0 + wave * 16 + half * 8;
#pragma unroll
    for (int t = 0; t < 4; ++t) {
        const int n = nW0 + t * 16 + l;
        const int h = n >> 8;          // head
        const int d = n & 255;         // 0..127 k_nope, 128..255 value
        const int c = (d < 128) ? (32 + h) : (64 + h);
        const int dd = d & 127;
#pragma unroll
        for (int v = 0; v < 8; ++v) {
            const int r = row + v;
            const int b = r >> 12, s = r & 4095;
            out[((size_t)(b * 96 + c) * 4096 + s) * 192 + dd] = acc[t][v];
        }
    }
}

// ---------------- RMSNorm: one 256-thread block per row --------------------
template<int L, int INSTRIDE, int OUTSTRIDE>
__global__ void __launch_bounds__(256) rmsnorm_k(
    const float* __restrict__ in, const float* __restrict__ w,
    float* __restrict__ out)
{
    __shared__ float red[8];
    const int row = blockIdx.x;
    const int tid = threadIdx.x;
    const float* x = in + (size_t)row * INSTRIDE;
    float* y = out + (size_t)row * OUTSTRIDE;

    float vals[L / 256];
    float ss = 0.f;
#pragma unroll
    for (int i = 0; i < L / 256; ++i) {
        vals[i] = x[tid + i * 256];
        ss += vals[i] * vals[i];
    }
    // wave32 reduce
#pragma unroll
    for (int off = 16; off > 0; off >>= 1)
        ss += __shfl_down(ss, off, 32);
    if ((tid & 31) == 0) red[tid >> 5] = ss;
    __syncthreads();
    if (tid == 0) {
        float tot = 0.f;
#pragma unroll
        for (int i = 0; i < 8; ++i) tot += red[i];
        red[0] = rsqrtf(tot * (1.0f / (float)L) + 1e-6f);
    }
    __syncthreads();
    const float rs = red[0];
#pragma unroll
    for (int i = 0; i < L / 256; ++i)
        y[tid + i * 256] = w[tid + i * 256] * vals[i] * rs;
}

// ---------------- k_pe broadcast (deepseek shuffle) + value pad zeros ------
__global__ void __launch_bounds__(64) kpe_vpad(
    const float* __restrict__ ckv, float* __restrict__ out)
{
    const int t = blockIdx.x;          // token 0..8191
    const int j = threadIdx.x;         // 0..63 (output rope dim)
    const int b = t >> 12, s = t & 4095;
    const int src = 2 * (j & 31) + (j >> 5);   // inverse deepseek transpose
    const float val = ckv[(size_t)t * 576 + 512 + src];
#pragma unroll 4
    for (int h = 0; h < 32; ++h) {
        out[((size_t)(b * 96 + 32 + h) * 4096 + s) * 192 + 128 + j] = val;  // key rope
        out[((size_t)(b * 96 + 64 + h) * 4096 + s) * 192 + 128 + j] = 0.f;  // v pad
    }
}

extern "C" void kernel_launch(void* const* d_in, const int* in_sizes, int n_in,
                              void* d_out, int out_size, void* d_ws, size_t ws_size,
                              hipStream_t stream) {
    const float* hidden  = (const float*)d_in[0];   // (2,4096,4096)
    const float* q_a_w   = (const float*)d_in[1];   // (1536,4096)
    const float* q_b_w   = (const float*)d_in[2];   // (6144,1536)
    const float* kv_a_w  = (const float*)d_in[3];   // (576,4096)
    const float* kv_b_w  = (const float*)d_in[4];   // (8192,512)
    const float* q_a_ln  = (const float*)d_in[5];   // (1536,)
    const float* kv_a_ln = (const float*)d_in[6];   // (512,)
    float* out = (float*)d_out;                     // (2,96,4096,192)

    float* q_mid = (float*)d_ws;                    // 8192*1536
    float* ckv   = q_mid + (size_t)8192 * 1536;     // 8192*576
    float* ckv_n = ckv   + (size_t)8192 * 576;      // 8192*512

    const dim3 blk(256);
    gemm_qa <<<dim3(1536 / BN, 8192 / BM), blk, 0, stream>>>(hidden, q_a_w, q_mid);
    gemm_kva<<<dim3( 576 / BN, 8192 / BM), blk, 0, stream>>>(hidden, kv_a_w, ckv);
    rmsnorm_k<1536, 1536, 1536><<<8192, 256, 0, stream>>>(q_mid, q_a_ln, q_mid);
    rmsnorm_k< 512,  576,  512><<<8192, 256, 0, stream>>>(ckv, kv_a_ln, ckv_n);
    gemm_qb <<<dim3(6144 / BN, 8192 / BM), blk, 0, stream>>>(q_mid, q_b_w, out);
    gemm_kvb<<<dim3(8192 / BN, 8192 / BM), blk, 0, stream>>>(ckv_n, kv_b_w, out);
    kpe_vpad<<<8192, 64, 0, stream>>>(ckv, out);
}